// GATv2Conv_16037407884002
// MI455X (gfx1250) — compile-verified
//
#include <hip/hip_runtime.h>
#include <hip/hip_bf16.h>
#include <math.h>

// ---------------------------------------------------------------------------
// GAT layer: h = x@W (fp32 WMMA), per-edge attention logits, segment softmax
// keyed by destination node, weighted scatter-add.  H=4 heads, C=16, IN=128.
// h (25.6MB) fits in MI455X 192MB L2, so edge gathers/atomics are L2-resident.
// ---------------------------------------------------------------------------

typedef __attribute__((ext_vector_type(2))) float v2f;
typedef __attribute__((ext_vector_type(8))) float v8f;

#define NEG_SLOPE 0.2f
#define EPS_F     1e-16f

__device__ __forceinline__ float lrelu(float v) {
    return v > 0.0f ? v : NEG_SLOPE * v;
}

// float atomic-max via signed/unsigned int atomics (monotone bit trick)
__device__ __forceinline__ void atomicMaxF(float* addr, float v) {
    if (v >= 0.0f) {
        atomicMax((int*)addr, __float_as_int(v));
    } else {
        atomicMin((unsigned int*)addr, (unsigned int)__float_as_int(v));
    }
}

// ---------------------------------------------------------------------------
// K1: h[N,64] = x[N,128] @ W[128,64] using V_WMMA_F32_16X16X4_F32.
// Block = 256 threads = 8 waves; block computes a 32x64 tile of h.
// Wave w: row-tile = w>>2 (0/1), col-tile = w&3 (0..3); K-loop of 32 WMMAs.
// LDS: X tile padded to stride 132, W padded to stride 65 (bank spread).
// ---------------------------------------------------------------------------
__global__ __launch_bounds__(256)
void gat_gemm_wmma(const float* __restrict__ x, const float* __restrict__ W,
                   float* __restrict__ h, int N)
{
    __shared__ float XT[32 * 132];   // 16896 B
    __shared__ float WT[128 * 65];   // 33280 B

    const int tid = threadIdx.x;
    const int r0  = blockIdx.x * 32;

    // Stage W [128][64] -> LDS (stride 65)
    for (int idx = tid; idx < 128 * 64; idx += 256) {
        int k = idx >> 6, n = idx & 63;
        WT[k * 65 + n] = W[idx];
    }
    // Stage X tile [32][128] -> LDS (stride 132), zero-pad OOB rows
    for (int idx = tid; idx < 32 * 128; idx += 256) {
        int r = idx >> 7, k = idx & 127;
        int row = r0 + r;
        XT[r * 132 + k] = (row < N) ? x[(size_t)row * 128 + k] : 0.0f;
    }
    __syncthreads();

    const int wave = tid >> 5;
    const int lane = tid & 31;
    const int rt   = wave >> 2;      // 0..1  -> rows rt*16..+15
    const int ct   = wave & 3;       // 0..3  -> cols ct*16..+15
    const int half = lane >> 4;      // 0: K={0,1}, 1: K={2,3} within a step
    const int mm   = lane & 15;      // M (for A) / N (for B)

    // A operand base: row (rt*16+mm) of X tile, +2 K when in upper half-wave
    const float* xrow = &XT[(rt * 16 + mm) * 132 + half * 2];
    // B operand base: W row (half*2), column ct*16+mm
    const float* wcol = &WT[(half * 2) * 65 + ct * 16 + mm];

    v8f acc = {0.f, 0.f, 0.f, 0.f, 0.f, 0.f, 0.f, 0.f};

#pragma unroll
    for (int k0 = 0; k0 < 128; k0 += 4) {
        v2f a, b;
        a.x = xrow[k0];
        a.y = xrow[k0 + 1];
        b.x = wcol[(k0    ) * 65];
        b.y = wcol[(k0 + 1) * 65];
        // 8 args: (neg_a, A, neg_b, B, c_mod, C, reuse_a, reuse_b)
        acc = __builtin_amdgcn_wmma_f32_16x16x4_f32(
            false, a, false, b, (short)0, acc, false, false);
    }

    // C/D layout: VGPR g holds M=g (lanes 0-15) and M=g+8 (lanes 16-31)
    const int rowBase = r0 + rt * 16;
#pragma unroll
    for (int g = 0; g < 8; ++g) {
        int m  = g + half * 8;
        int rr = rowBase + m;
        if (rr < N) h[(size_t)rr * 64 + ct * 16 + mm] = acc[g];
    }
}

// ---------------------------------------------------------------------------
// K2: self-loop logit s[i,h] = sum_c lrelu(2*h[i,h,c]) * att[h,c];
//     initialize amax with it (every node has a self loop -> non-empty seg).
// ---------------------------------------------------------------------------
__global__ __launch_bounds__(256)
void gat_self_logit(const float* __restrict__ h, const float* __restrict__ att,
                    float* __restrict__ selfL, float* __restrict__ amax, int N)
{
    int t = blockIdx.x * 256 + threadIdx.x;
    if (t >= N * 4) return;
    int i = t >> 2, hh = t & 3;
    const float4* hv = (const float4*)(h + (size_t)i * 64 + hh * 16);
    const float4* ap = (const float4*)(att + hh * 16);
    float s = 0.0f;
#pragma unroll
    for (int q = 0; q < 4; ++q) {
        float4 a4 = ap[q], h4 = hv[q];
        s += lrelu(2.0f * h4.x) * a4.x;
        s += lrelu(2.0f * h4.y) * a4.y;
        s += lrelu(2.0f * h4.z) * a4.z;
        s += lrelu(2.0f * h4.w) * a4.w;
    }
    selfL[t] = s;
    amax[t]  = s;
}

// ---------------------------------------------------------------------------
// K3: per-edge logit; store to ws and atomic-max into amax[row].
// Thread = (edge, head). Gathers hit L2 (h is L2-resident).
// ---------------------------------------------------------------------------
__global__ __launch_bounds__(256)
void gat_edge_logit(const float* __restrict__ h, const int* __restrict__ ei,
                    const float* __restrict__ att, float* __restrict__ logit,
                    float* __restrict__ amax, int E)
{
    int t = blockIdx.x * 256 + threadIdx.x;
    if (t >= E * 4) return;
    int e = t >> 2, hh = t & 3;
    int row = ei[e];       // target
    int col = ei[E + e];   // source
    const float4* hi = (const float4*)(h + (size_t)row * 64 + hh * 16);
    const float4* hj = (const float4*)(h + (size_t)col * 64 + hh * 16);
    const float4* ap = (const float4*)(att + hh * 16);
    float s = 0.0f;
#pragma unroll
    for (int q = 0; q < 4; ++q) {
        float4 a4 = ap[q], i4 = hi[q], j4 = hj[q];
        s += lrelu(i4.x + j4.x) * a4.x;
        s += lrelu(i4.y + j4.y) * a4.y;
        s += lrelu(i4.z + j4.z) * a4.z;
        s += lrelu(i4.w + j4.w) * a4.w;
    }
    logit[t] = s;
    atomicMaxF(&amax[(size_t)row * 4 + hh], s);
}

// ---------------------------------------------------------------------------
// K4: denom[i,h] = exp(selfL - amax)   (self-loop contribution to softmax sum)
// ---------------------------------------------------------------------------
__global__ __launch_bounds__(256)
void gat_denom_init(const float* __restrict__ selfL,
                    const float* __restrict__ amax,
                    float* __restrict__ denom, int N)
{
    int t = blockIdx.x * 256 + threadIdx.x;
    if (t >= N * 4) return;
    denom[t] = expf(selfL[t] - amax[t]);
}

// ---------------------------------------------------------------------------
// K5: ea = exp(logit - amax[row]); overwrite logit with ea; atomicAdd denom.
// Re-reads only 4B/thread instead of re-gathering 512B/edge.
// ---------------------------------------------------------------------------
__global__ __launch_bounds__(256)
void gat_edge_exp(const int* __restrict__ ei, const float* __restrict__ amax,
                  float* __restrict__ logit, float* __restrict__ denom, int E)
{
    int t = blockIdx.x * 256 + threadIdx.x;
    if (t >= E * 4) return;
    int e = t >> 2, hh = t & 3;
    int row = ei[e];
    float ea = expf(logit[t] - amax[(size_t)row * 4 + hh]);
    logit[t] = ea;
    atomicAdd(&denom[(size_t)row * 4 + hh], ea);
}

// ---------------------------------------------------------------------------
// K6: out[i] = bias + (exp(selfL-amax)/(denom+eps)) * h[i]   (self term)
// Also serves as the deterministic initialization of d_out.
// ---------------------------------------------------------------------------
__global__ __launch_bounds__(256)
void gat_out_init(const float* __restrict__ h, const float* __restrict__ selfL,
                  const float* __restrict__ amax, const float* __restrict__ denom,
                  const float* __restrict__ bias, float* __restrict__ out, int N)
{
    int t = blockIdx.x * 256 + threadIdx.x;
    if (t >= N * 4) return;
    int i = t >> 2, hh = t & 3;
    float coef = expf(selfL[t] - amax[t]) / (denom[t] + EPS_F);
    const float4* hv = (const float4*)(h + (size_t)i * 64 + hh * 16);
    const float4* bp = (const float4*)(bias + hh * 16);
    float4*       op = (float4*)(out + (size_t)i * 64 + hh * 16);
#pragma unroll
    for (int q = 0; q < 4; ++q) {
        float4 h4 = hv[q], b4 = bp[q], o4;
        o4.x = b4.x + coef * h4.x;
        o4.y = b4.y + coef * h4.y;
        o4.z = b4.z + coef * h4.z;
        o4.w = b4.w + coef * h4.w;
        op[q] = o4;
    }
}

// ---------------------------------------------------------------------------
// K7: weighted message scatter: out[row] += (ea/(denom[row]+eps)) * h[col].
// Non-returning global_atomic_add_f32 -> STOREcnt, no wave stall; L2-resident.
// ---------------------------------------------------------------------------
__global__ __launch_bounds__(256)
void gat_edge_scatter(const float* __restrict__ h, const int* __restrict__ ei,
                      const float* __restrict__ logit,
                      const float* __restrict__ denom,
                      float* __restrict__ out, int E)
{
    int t = blockIdx.x * 256 + threadIdx.x;
    if (t >= E * 4) return;
    int e = t >> 2, hh = t & 3;
    int row = ei[e];
    int col = ei[E + e];
    float coef = logit[t] / (denom[(size_t)row * 4 + hh] + EPS_F);
    const float* hj = h + (size_t)col * 64 + hh * 16;
    float*       op = out + (size_t)row * 64 + hh * 16;
#pragma unroll
    for (int c = 0; c < 16; ++c) {
        atomicAdd(&op[c], coef * hj[c]);
    }
}

// ---------------------------------------------------------------------------
extern "C" void kernel_launch(void* const* d_in, const int* in_sizes, int n_in,
                              void* d_out, int out_size, void* d_ws, size_t ws_size,
                              hipStream_t stream)
{
    const float* x    = (const float*)d_in[0];   // [N,128]
    const int*   ei   = (const int*)  d_in[1];   // [2,E]
    const float* W    = (const float*)d_in[2];   // [128,64]
    const float* att  = (const float*)d_in[3];   // [1,4,16]
    const float* bias = (const float*)d_in[4];   // [64]
    float*       out  = (float*)d_out;           // [N,64]

    const int N = in_sizes[0] / 128;
    const int E = in_sizes[1] / 2;

    // Workspace layout (floats): h[N*64] | logit/ea[E*4] | selfL[N*4] |
    //                            amax[N*4] | denom[N*4]   (~56 MB total)
    float* h     = (float*)d_ws;
    float* logit = h     + (size_t)N * 64;
    float* selfL = logit + (size_t)E * 4;
    float* amax  = selfL + (size_t)N * 4;
    float* denom = amax  + (size_t)N * 4;

    dim3 blk(256);
    int gGemm = (N + 31) / 32;
    int gNode = (N * 4 + 255) / 256;
    int gEdge = (int)(((long long)E * 4 + 255) / 256);

    gat_gemm_wmma  <<<gGemm, blk, 0, stream>>>(x, W, h, N);
    gat_self_logit <<<gNode, blk, 0, stream>>>(h, att, selfL, amax, N);
    gat_edge_logit <<<gEdge, blk, 0, stream>>>(h, ei, att, logit, amax, E);
    gat_denom_init <<<gNode, blk, 0, stream>>>(selfL, amax, denom, N);
    gat_edge_exp   <<<gEdge, blk, 0, stream>>>(ei, amax, logit, denom, E);
    gat_out_init   <<<gNode, blk, 0, stream>>>(h, selfL, amax, denom, bias, out, N);
    gat_edge_scatter<<<gEdge, blk, 0, stream>>>(h, ei, logit, denom, out, E);
}